// GCNEncoder_78812649882125
// MI455X (gfx1250) — compile-verified
//
#include <hip/hip_runtime.h>
#include <hip/hip_bf16.h>

typedef __attribute__((ext_vector_type(2))) float v2f;
typedef __attribute__((ext_vector_type(8))) float v8f;

#define HID 64

// ---------------- utility kernels ----------------

__global__ void k_fill(float* __restrict__ p, float v, long long n) {
  long long i = (long long)blockIdx.x * blockDim.x + threadIdx.x;
  if (i < n) p[i] = v;
}

__global__ void k_deg_count(const int* __restrict__ dst, float* __restrict__ deg, int E) {
  int e = blockIdx.x * blockDim.x + threadIdx.x;
  if (e < E) unsafeAtomicAdd(&deg[dst[e]], 1.0f);
}

__global__ void k_rsqrt_inplace(float* __restrict__ p, int n) {
  int i = blockIdx.x * blockDim.x + threadIdx.x;
  if (i < n) p[i] = rsqrtf(p[i]);
}

// ---------------- WMMA GEMM: C[nrows x 64] = A[nrows x K] * B[K x 64] --------
// One wave computes one 16x16 tile of C using V_WMMA_F32_16X16X4_F32.
// A-frag (16x4 f32, 2 VGPRs): lanes 0-15 hold {K=0,K=1} of row M=lane,
//                             lanes 16-31 hold {K=2,K=3} of row M=lane-16.
// B-frag (4x16 f32, 2 VGPRs): lanes 0-15 hold N=lane of rows {K=0,K=1},
//                             lanes 16-31 hold N=lane-16 of rows {K=2,K=3}.
// C/D (16x16 f32, 8 VGPRs): VGPR v: lanes 0-15 -> M=v, lanes 16-31 -> M=v+8.
//
// A row index is clamped (no per-lane exec predication); C rows are written
// unguarded, so C must be padded up to a multiple of 16 rows (workspace only).
template <int K>
__global__ void k_gemm_wmma(const float* __restrict__ A, const float* __restrict__ B,
                            float* __restrict__ C, int nrows) {
  __shared__ float sB[K * HID];

  // cooperative copy of B (K x 64 floats) into LDS, float4 per transaction
  for (int i = threadIdx.x * 4; i < K * HID; i += blockDim.x * 4) {
    *(float4*)&sB[i] = *(const float4*)&B[i];
  }
  __syncthreads();

  const int wave   = (blockIdx.x * blockDim.x + threadIdx.x) >> 5;
  const int lane   = threadIdx.x & 31;
  const int tilesN = HID / 16;                 // 4 tiles across the 64 columns
  const int tileM  = wave / tilesN;
  const int tileN  = wave % tilesN;
  const int row0   = tileM * 16;
  const int col0   = tileN * 16;
  if (row0 >= nrows) return;                   // wave-uniform exit

  const int half = lane >> 4;                  // 0 -> K pair {0,1}; 1 -> {2,3}
  const int lrow = lane & 15;
  int arow = row0 + lrow;
  arow = arow < nrows ? arow : nrows - 1;      // clamp instead of predication

  const float* ap = A + (long long)arow * K + half * 2;   // 8B aligned
  const float* bp = sB + (half * 2) * HID + col0 + lrow;

  v8f acc = {};
  #pragma unroll 4
  for (int k = 0; k < K; k += 4) {
    v2f a = *(const v2f*)(ap + k);             // global_load_b64
    v2f b;
    b.x = bp[k * HID];                         // ds_load (pairs mergeable)
    b.y = bp[k * HID + HID];
    acc = __builtin_amdgcn_wmma_f32_16x16x4_f32(false, a, false, b,
                                                (short)0, acc, false, false);
  }

  float* cp = C + (long long)(row0 + half * 8) * HID + col0 + lrow;
  #pragma unroll
  for (int v = 0; v < 8; ++v) {
    cp[(long long)v * HID] = acc[v];           // unguarded: C padded to 16 rows
  }
}

// ---------------- edge scatter: agg[dst] += h[src] * dis[src]*dis[dst] -------
// One wave per edge; each lane moves 2 floats (64 cols / 32 lanes).
__global__ void k_edge_agg(const int* __restrict__ src, const int* __restrict__ dst,
                           const float* __restrict__ dis, const float* __restrict__ h,
                           float* __restrict__ agg, int E) {
  int e    = (blockIdx.x * blockDim.x + threadIdx.x) >> 5;
  int lane = threadIdx.x & 31;
  if (e >= E) return;
  int s = src[e];
  int d = dst[e];
  float nrm = dis[s] * dis[d];
  const float2* hp = (const float2*)(h + (long long)s * HID);
  float2 v = hp[lane];
  float* ap = agg + (long long)d * HID + 2 * lane;
  unsafeAtomicAdd(ap,     v.x * nrm);
  unsafeAtomicAdd(ap + 1, v.y * nrm);
}

// ---------------- self-loop + bias: agg += h * dis^2 + bias ------------------
__global__ void k_selfloop_bias(const float* __restrict__ h, const float* __restrict__ dis,
                                const float* __restrict__ bias, float* __restrict__ agg,
                                int n) {
  long long i = (long long)blockIdx.x * blockDim.x + threadIdx.x;
  if (i >= (long long)n * HID) return;
  int r = (int)(i >> 6);
  int c = (int)(i & 63);
  float ds = dis[r];
  agg[i] += h[i] * ds * ds + bias[c];
}

// ---------------- BatchNorm stats: per-column sum / sumsq --------------------
__global__ void k_bn_stats(const float* __restrict__ g, float* __restrict__ sum,
                           float* __restrict__ sumsq, int n) {
  int c = threadIdx.x;                         // 0..63
  float s = 0.0f, s2 = 0.0f;
  for (int r = blockIdx.x; r < n; r += gridDim.x) {
    float v = g[(long long)r * HID + c];
    s  += v;
    s2 += v * v;
  }
  unsafeAtomicAdd(&sum[c], s);
  unsafeAtomicAdd(&sumsq[c], s2);
}

// ---------------- BN normalize + affine + ReLU -------------------------------
__global__ void k_bn_apply_relu(const float* __restrict__ g, const float* __restrict__ sum,
                                const float* __restrict__ sumsq, const float* __restrict__ w,
                                const float* __restrict__ b, float* __restrict__ out, int n) {
  long long i = (long long)blockIdx.x * blockDim.x + threadIdx.x;
  if (i >= (long long)n * HID) return;
  int c = (int)(i & 63);
  float invN  = 1.0f / (float)n;
  float mean  = sum[c] * invN;
  float var   = sumsq[c] * invN - mean * mean;
  float v = (g[i] - mean) * rsqrtf(var + 1e-5f) * w[c] + b[c];
  out[i] = v > 0.0f ? v : 0.0f;
}

// ---------------- host-side orchestration ------------------------------------

extern "C" void kernel_launch(void* const* d_in, const int* in_sizes, int n_in,
                              void* d_out, int out_size, void* d_ws, size_t ws_size,
                              hipStream_t stream) {
  const float* x   = (const float*)d_in[0];
  const int*   ei  = (const int*)d_in[1];
  const float* W1  = (const float*)d_in[2];
  const float* b1  = (const float*)d_in[3];
  const float* bnw = (const float*)d_in[4];
  const float* bnb = (const float*)d_in[5];
  const float* W2  = (const float*)d_in[6];
  const float* b2  = (const float*)d_in[7];

  const int IN = 128;
  const int N  = in_sizes[0] / IN;
  const int E  = in_sizes[1] / 2;
  const int* src = ei;
  const int* dst = ei + E;

  const long long Npad = ((long long)N + 15) & ~15LL;   // GEMM stores 16-row tiles

  float* ws    = (float*)d_ws;
  float* dis   = ws;                       // [N]
  float* H     = dis + N;                  // [Npad*64]
  float* G     = H + Npad * HID;           // [Npad*64]
  float* bnsum = G + Npad * HID;           // [64]
  float* bnsq  = bnsum + HID;              // [64]
  float* out   = (float*)d_out;            // [N*64]

  const long long NH = (long long)N * HID;
  const int T = 256;
  auto cdiv = [](long long a, long long b) { return (int)((a + b - 1) / b); };

  // 1) degrees (self-loop included) -> dis = rsqrt(deg)
  k_fill<<<cdiv(N, T), T, 0, stream>>>(dis, 1.0f, N);
  k_deg_count<<<cdiv(E, T), T, 0, stream>>>(dst, dis, E);
  k_rsqrt_inplace<<<cdiv(N, T), T, 0, stream>>>(dis, N);

  // 2) H = x @ W1   (WMMA, K=128)
  {
    int waves = cdiv(N, 16) * (HID / 16);
    k_gemm_wmma<128><<<cdiv((long long)waves * 32, T), T, 0, stream>>>(x, W1, H, N);
  }

  // 3) G = scatter(H) + selfloop + b1
  k_fill<<<cdiv(NH, T), T, 0, stream>>>(G, 0.0f, NH);
  k_edge_agg<<<cdiv((long long)E * 32, T), T, 0, stream>>>(src, dst, dis, H, G, E);
  k_selfloop_bias<<<cdiv(NH, T), T, 0, stream>>>(H, dis, b1, G, N);

  // 4) BatchNorm (training stats) + ReLU  -> H
  k_fill<<<1, 2 * HID, 0, stream>>>(bnsum, 0.0f, 2 * HID);
  k_bn_stats<<<512, HID, 0, stream>>>(G, bnsum, bnsq, N);
  k_bn_apply_relu<<<cdiv(NH, T), T, 0, stream>>>(G, bnsum, bnsq, bnw, bnb, H, N);

  // 5) G = H @ W2   (WMMA, K=64)
  {
    int waves = cdiv(N, 16) * (HID / 16);
    k_gemm_wmma<64><<<cdiv((long long)waves * 32, T), T, 0, stream>>>(H, W2, G, N);
  }

  // 6) out = scatter(G) + selfloop + b2
  k_fill<<<cdiv(NH, T), T, 0, stream>>>(out, 0.0f, NH);
  k_edge_agg<<<cdiv((long long)E * 32, T), T, 0, stream>>>(src, dst, dis, G, out, E);
  k_selfloop_bias<<<cdiv(NH, T), T, 0, stream>>>(G, dis, b2, out, N);
}